// DepthAwareConv2d_53369263620275
// MI455X (gfx1250) — compile-verified
//
#include <hip/hip_runtime.h>

typedef float v2f __attribute__((ext_vector_type(2)));
typedef float v8f __attribute__((ext_vector_type(8)));

#define NB   4
#define CIN  128
#define OOUT 256
#define HH   128
#define WW   128
#define HWs  (HH * WW)

// Repack weight (O,C,3,3) -> wpk[p][c/2][o] as float2 = (w[o][c0][p], w[o][c0+1][p])
// so the WMMA f32 16x16x4 A-operand load is one coalesced b64 per lane.
__global__ __launch_bounds__(256) void repack_w_kernel(const float* __restrict__ w,
                                                       float* __restrict__ wpk) {
    int idx = blockIdx.x * 256 + threadIdx.x;           // O*C*9 = 294912 total
    if (idx >= OOUT * CIN * 9) return;
    int p = idx % 9;
    int c = (idx / 9) % CIN;
    int o = idx / (9 * CIN);
    // dst float index: ((p*(C/2) + c/2)*O + o)*2 + (c&1)
    wpk[((((size_t)p * (CIN / 2) + (c >> 1)) * OOUT + o) << 1) + (c & 1)] = w[idx];
}

// Implicit-GEMM depth-modulated conv. Block = 256 o-channels x 64 pixels of one row.
// 8 waves: 4 along M (64 o each = 4 tiles), 2 along N (32 px each = 2 tiles).
__global__ __launch_bounds__(256) void depth_conv_wmma_kernel(
    const float* __restrict__ x, const float* __restrict__ depth,
    const float* __restrict__ wpk, const float* __restrict__ bias,
    float* __restrict__ out) {
    const int pb   = blockIdx.x;            // 1024 blocks = 4 n * 128 rows * 2 xblk
    const int n    = pb >> 8;
    const int rem  = pb & 255;
    const int y    = rem >> 1;
    const int x0   = (rem & 1) * 64;

    const int tid  = threadIdx.x;
    const int lane = tid & 31;
    const int wave = tid >> 5;
    const int mw   = wave & 3;              // M wave coord (0..3)
    const int nw   = wave >> 2;             // N wave coord (0..1)
    const int nn   = lane & 15;             // lane%16: M row for A, N col for B/C
    const int kh   = lane >> 4;             // lane/16: K-half selector

    const int o_wave = mw * 64;             // 4 M-tiles: o_wave + {0,16,32,48}
    const int px0    = x0 + nw * 32;        // 2 N-tiles: px0, px0+16

    const float* __restrict__ xn = x + (size_t)n * CIN * HWs;
    const float* __restrict__ dn = depth + (size_t)n * HWs;
    const v2f*   __restrict__ wv = (const v2f*)wpk;

    v8f acc[4][2];
#pragma unroll
    for (int m = 0; m < 4; ++m) {
        acc[m][0] = (v8f){0.f, 0.f, 0.f, 0.f, 0.f, 0.f, 0.f, 0.f};
        acc[m][1] = (v8f){0.f, 0.f, 0.f, 0.f, 0.f, 0.f, 0.f, 0.f};
    }

#pragma unroll
    for (int p = 0; p < 9; ++p) {
        const int di = p / 3 - 1;
        const int dj = p % 3 - 1;
        const int yy = y + di;
        if (yy < 0 || yy >= HH) continue;   // wave-uniform: EXEC stays all-ones

        // Per-lane pixel coords for the two N-tiles; clamp for safe x loads,
        // zero depth kills padded taps (both patches are zero-padded together).
        const int xx0 = px0 + nn + dj;
        const int xx1 = xx0 + 16;
        const int xc0 = min(max(xx0, 0), WW - 1);
        const int xc1 = min(max(xx1, 0), WW - 1);
        const float d0 = (xx0 >= 0 && xx0 < WW) ? dn[yy * WW + xx0] : 0.0f;
        const float d1 = (xx1 >= 0 && xx1 < WW) ? dn[yy * WW + xx1] : 0.0f;

        // B source: channels c0+2kh, c0+2kh+1 at row yy
        const float* __restrict__ xp = xn + (size_t)(2 * kh) * HWs + (size_t)yy * WW;
        // A source: wpk[p][c0/2 + kh][o]
        const v2f* __restrict__ wrow =
            wv + ((size_t)p * (CIN / 2) + kh) * OOUT + (o_wave + nn);

#pragma unroll 2
        for (int c0 = 0; c0 < CIN; c0 += 4) {
            v2f b0, b1;
            b0.x = xp[xc0];
            b0.y = xp[HWs + xc0];
            b1.x = xp[xc1];
            b1.y = xp[HWs + xc1];
            // fold the depth modulation into B (matches reference x_patch*d_patch)
            b0.x *= d0; b0.y *= d0;
            b1.x *= d1; b1.y *= d1;
#pragma unroll
            for (int m = 0; m < 4; ++m) {
                v2f a = wrow[m * 16];
                acc[m][0] = __builtin_amdgcn_wmma_f32_16x16x4_f32(
                    false, a, false, b0, (short)0, acc[m][0], false, false);
                acc[m][1] = __builtin_amdgcn_wmma_f32_16x16x4_f32(
                    false, a, false, b1, (short)0, acc[m][1], false, false);
            }
            xp   += 4 * HWs;                // advance 4 channels
            wrow += 2 * OOUT;               // advance 2 c-pairs
        }
    }

    // Epilogue: C/D layout -> row o = tile + 8*kh + v, col px = tile + nn.
#pragma unroll
    for (int m = 0; m < 4; ++m) {
        const int o_row = o_wave + m * 16 + 8 * kh;
#pragma unroll
        for (int t = 0; t < 2; ++t) {
            const int px = px0 + t * 16 + nn;
#pragma unroll
            for (int v = 0; v < 8; ++v) {
                const int o = o_row + v;
                out[(((size_t)n * OOUT + o) * HH + y) * WW + px] =
                    acc[m][t][v] + bias[o];
            }
        }
    }
}

extern "C" void kernel_launch(void* const* d_in, const int* in_sizes, int n_in,
                              void* d_out, int out_size, void* d_ws, size_t ws_size,
                              hipStream_t stream) {
    const float* x      = (const float*)d_in[0];
    const float* depth  = (const float*)d_in[1];
    // d_in[2] = camera_params (unused by reference)
    const float* weight = (const float*)d_in[3];
    const float* bias   = (const float*)d_in[4];
    float* out = (float*)d_out;
    float* wpk = (float*)d_ws;              // needs O*C*9*4 = 1,179,648 bytes

    repack_w_kernel<<<(OOUT * CIN * 9 + 255) / 256, 256, 0, stream>>>(weight, wpk);
    depth_conv_wmma_kernel<<<NB * HH * 2, 256, 0, stream>>>(x, depth, wpk, bias, out);
}